// MultiHeadAttention_27324581937550
// MI455X (gfx1250) — compile-verified
//
#include <hip/hip_runtime.h>
#include <cstddef>
#include <math.h>

// ---------------------------------------------------------------------------
// MultiHeadAttention on MI455X (gfx1250), fp32 WMMA path.
// Memory-bound problem (probs tensor ~403MB dominates) -> keep fp32 accuracy
// and use V_WMMA_F32_16X16X4_F32 for every GEMM.
// ---------------------------------------------------------------------------

typedef __attribute__((ext_vector_type(2))) float v2f;
typedef __attribute__((ext_vector_type(8))) float v8f;

static constexpr int Bn = 8;     // batch
static constexpr int E  = 768;   // D_EMBED == D_MODEL
static constexpr int S  = 1024;  // IMG*IMG
static constexpr int NH = 12;    // heads
static constexpr int DP = 64;    // depth

__device__ __forceinline__ v8f wmma_f32(v2f a, v2f b, v8f c) {
  // 8 args: (neg_a, A, neg_b, B, c_mod, C, reuse_a, reuse_b)
  return __builtin_amdgcn_wmma_f32_16x16x4_f32(
      false, a, false, b, (short)0, c, false, false);
}

// -------------------------------------------------------------------------
// Projection: Y[s,o] = sum_c X[b,c,s] * W[o,c] + bias[o]
//   x : [B, 768, 1024] (channel-major image, spatial flattened)
//   w : [768, 768] row-major [O,C]
//   out: transpose_out==0 -> [B*NH, S, DP]   (Q, V)
//        transpose_out==1 -> [B*NH, DP, S]   (K stored transposed for scores)
// One wave computes a 32x32 tile of [S=1024 x O=768].
// -------------------------------------------------------------------------
__global__ void proj_kernel(const float* __restrict__ x,
                            const float* __restrict__ w,
                            const float* __restrict__ bias,
                            float* __restrict__ out,
                            int transpose_out) {
  const int b     = blockIdx.y;
  const int wave  = threadIdx.x >> 5;
  const int lane  = threadIdx.x & 31;
  const int tile  = blockIdx.x * 4 + wave;   // 768 tiles: 32 (M/S) x 24 (N/O)
  const int tm    = tile & 31;
  const int tn    = tile >> 5;
  const int m0    = tm * 32, n0 = tn * 32;
  const int l15   = lane & 15;
  const int khalf = (lane >> 4) * 2;         // A/B K-pair select per lane half
  const int rhalf = (lane >> 4) * 8;         // C/D row offset per lane half

  const float* xb = x + (size_t)b * E * S;

  const v8f zero = {};
  v8f acc[2][2] = {{zero, zero}, {zero, zero}};

  for (int c0 = 0; c0 < E; c0 += 4) {
    v2f a[2], bb[2];
#pragma unroll
    for (int mi = 0; mi < 2; ++mi) {
      const int row = m0 + mi * 16 + l15;
      a[mi].x = xb[(size_t)(c0 + khalf)     * S + row];
      a[mi].y = xb[(size_t)(c0 + khalf + 1) * S + row];
    }
#pragma unroll
    for (int ni = 0; ni < 2; ++ni) {
      const int col = n0 + ni * 16 + l15;
      const float2 wv =
          *reinterpret_cast<const float2*>(w + (size_t)col * E + c0 + khalf);
      bb[ni].x = wv.x;
      bb[ni].y = wv.y;
    }
#pragma unroll
    for (int mi = 0; mi < 2; ++mi)
#pragma unroll
      for (int ni = 0; ni < 2; ++ni)
        acc[mi][ni] = wmma_f32(a[mi], bb[ni], acc[mi][ni]);
  }

#pragma unroll
  for (int mi = 0; mi < 2; ++mi) {
#pragma unroll
    for (int ni = 0; ni < 2; ++ni) {
      const int col = n0 + ni * 16 + l15;   // output channel o
      const float bv = bias[col];
      const int hh = col >> 6, dd = col & 63;
#pragma unroll
      for (int r = 0; r < 8; ++r) {
        const int srow = m0 + mi * 16 + rhalf + r;
        const float val = acc[mi][ni][r] + bv;
        size_t idx;
        if (transpose_out)
          idx = ((size_t)(b * NH + hh) * DP + dd) * S + srow;
        else
          idx = ((size_t)(b * NH + hh) * S + srow) * DP + dd;
        out[idx] = val;
      }
    }
  }
}

// -------------------------------------------------------------------------
// Scores: scores[q,k] = (1/8) * sum_d Q[q,d] * K[k,d]
//   q  : [BH, 1024, 64], kt : [BH, 64, 1024] -> scores : [BH, 1024, 1024]
// -------------------------------------------------------------------------
__global__ void scores_kernel(const float* __restrict__ q,
                              const float* __restrict__ kt,
                              float* __restrict__ scores) {
  const int bh    = blockIdx.y;
  const int wave  = threadIdx.x >> 5;
  const int lane  = threadIdx.x & 31;
  const int tile  = blockIdx.x * 4 + wave;   // 1024 tiles: 32 x 32
  const int tm    = tile & 31;
  const int tn    = tile >> 5;
  const int m0    = tm * 32, n0 = tn * 32;
  const int l15   = lane & 15;
  const int khalf = (lane >> 4) * 2;
  const int rhalf = (lane >> 4) * 8;

  const float* qb = q  + (size_t)bh * S * DP;
  const float* kb = kt + (size_t)bh * DP * S;

  const v8f zero = {};
  v8f acc[2][2] = {{zero, zero}, {zero, zero}};

  for (int c0 = 0; c0 < DP; c0 += 4) {
    v2f a[2], bb[2];
#pragma unroll
    for (int mi = 0; mi < 2; ++mi) {
      const int row = m0 + mi * 16 + l15;
      const float2 t =
          *reinterpret_cast<const float2*>(qb + (size_t)row * DP + c0 + khalf);
      a[mi].x = t.x;
      a[mi].y = t.y;
    }
#pragma unroll
    for (int ni = 0; ni < 2; ++ni) {
      const int col = n0 + ni * 16 + l15;
      bb[ni].x = kb[(size_t)(c0 + khalf)     * S + col];
      bb[ni].y = kb[(size_t)(c0 + khalf + 1) * S + col];
    }
#pragma unroll
    for (int mi = 0; mi < 2; ++mi)
#pragma unroll
      for (int ni = 0; ni < 2; ++ni)
        acc[mi][ni] = wmma_f32(a[mi], bb[ni], acc[mi][ni]);
  }

  const float scale = 0.125f;  // 1/sqrt(64)
#pragma unroll
  for (int mi = 0; mi < 2; ++mi) {
#pragma unroll
    for (int ni = 0; ni < 2; ++ni) {
      const int col = n0 + ni * 16 + l15;
#pragma unroll
      for (int r = 0; r < 8; ++r) {
        const int srow = m0 + mi * 16 + rhalf + r;
        scores[((size_t)bh * S + srow) * S + col] = acc[mi][ni][r] * scale;
      }
    }
  }
}

// -------------------------------------------------------------------------
// Row softmax over 1024 elements, in place. One 256-thread block per row.
// -------------------------------------------------------------------------
__global__ void softmax_kernel(float* __restrict__ probs) {
  float* p = probs + (size_t)blockIdx.x * S;
  const int t = threadIdx.x;

  float v[4];
  float mx = -3.402823466e38f;
#pragma unroll
  for (int i = 0; i < 4; ++i) {
    v[i] = p[t + i * 256];
    mx = fmaxf(mx, v[i]);
  }

  __shared__ float red[256];
  red[t] = mx;
  __syncthreads();
  for (int s = 128; s > 0; s >>= 1) {
    if (t < s) red[t] = fmaxf(red[t], red[t + s]);
    __syncthreads();
  }
  mx = red[0];
  __syncthreads();

  float sum = 0.0f;
#pragma unroll
  for (int i = 0; i < 4; ++i) {
    v[i] = expf(v[i] - mx);
    sum += v[i];
  }
  red[t] = sum;
  __syncthreads();
  for (int s = 128; s > 0; s >>= 1) {
    if (t < s) red[t] += red[t + s];
    __syncthreads();
  }
  const float inv = 1.0f / red[0];
#pragma unroll
  for (int i = 0; i < 4; ++i) p[t + i * 256] = v[i] * inv;
}

// -------------------------------------------------------------------------
// Context: ctx[q,d] = sum_k probs[q,k] * V[k,d], merged-head output layout
//   probs : [BH, 1024, 1024], v : [BH, 1024, 64] -> ctx : [B, 1024, 768]
// -------------------------------------------------------------------------
__global__ void ctx_kernel(const float* __restrict__ probs,
                           const float* __restrict__ v,
                           float* __restrict__ ctx) {
  const int bh    = blockIdx.y;
  const int b     = bh / NH;
  const int hh    = bh % NH;
  const int wave  = threadIdx.x >> 5;
  const int lane  = threadIdx.x & 31;
  const int tile  = blockIdx.x * 4 + wave;   // 64 tiles: 32 (M/q) x 2 (N/d)
  const int tm    = tile & 31;
  const int tn    = tile >> 5;
  const int m0    = tm * 32, n0 = tn * 32;
  const int l15   = lane & 15;
  const int khalf = (lane >> 4) * 2;
  const int rhalf = (lane >> 4) * 8;

  const float* pb = probs + (size_t)bh * S * S;
  const float* vb = v     + (size_t)bh * S * DP;

  const v8f zero = {};
  v8f acc[2][2] = {{zero, zero}, {zero, zero}};

  for (int c0 = 0; c0 < S; c0 += 4) {
    v2f a[2], bb[2];
#pragma unroll
    for (int mi = 0; mi < 2; ++mi) {
      const int row = m0 + mi * 16 + l15;
      const float2 t =
          *reinterpret_cast<const float2*>(pb + (size_t)row * S + c0 + khalf);
      a[mi].x = t.x;
      a[mi].y = t.y;
    }
#pragma unroll
    for (int ni = 0; ni < 2; ++ni) {
      const int col = n0 + ni * 16 + l15;
      bb[ni].x = vb[(size_t)(c0 + khalf)     * DP + col];
      bb[ni].y = vb[(size_t)(c0 + khalf + 1) * DP + col];
    }
#pragma unroll
    for (int mi = 0; mi < 2; ++mi)
#pragma unroll
      for (int ni = 0; ni < 2; ++ni)
        acc[mi][ni] = wmma_f32(a[mi], bb[ni], acc[mi][ni]);
  }

#pragma unroll
  for (int mi = 0; mi < 2; ++mi) {
#pragma unroll
    for (int ni = 0; ni < 2; ++ni) {
      const int col = n0 + ni * 16 + l15;   // depth index within head
#pragma unroll
      for (int r = 0; r < 8; ++r) {
        const int srow = m0 + mi * 16 + rhalf + r;
        ctx[((size_t)b * S + srow) * E + hh * DP + col] = acc[mi][ni][r];
      }
    }
  }
}

// -------------------------------------------------------------------------
// Final FC: out[b,o,s] = sum_c ctx[b,s,c] * fc_w[o,c] + fc_b[o]
//   ctx : [B, 1024, 768], w : [768, 768] -> out : [B, 768, 1024]
// M = o (so stores are s-contiguous across lanes).
// -------------------------------------------------------------------------
__global__ void fc_kernel(const float* __restrict__ ctx,
                          const float* __restrict__ w,
                          const float* __restrict__ bias,
                          float* __restrict__ out) {
  const int b     = blockIdx.y;
  const int wave  = threadIdx.x >> 5;
  const int lane  = threadIdx.x & 31;
  const int tile  = blockIdx.x * 4 + wave;   // 768 tiles: 24 (M/o) x 32 (N/s)
  const int tn    = tile & 31;
  const int tm    = tile >> 5;
  const int m0    = tm * 32, n0 = tn * 32;
  const int l15   = lane & 15;
  const int khalf = (lane >> 4) * 2;
  const int rhalf = (lane >> 4) * 8;

  const float* cb = ctx + (size_t)b * S * E;

  const v8f zero = {};
  v8f acc[2][2] = {{zero, zero}, {zero, zero}};

  for (int c0 = 0; c0 < E; c0 += 4) {
    v2f a[2], bb[2];
#pragma unroll
    for (int mi = 0; mi < 2; ++mi) {
      const int row = m0 + mi * 16 + l15;   // o
      const float2 t =
          *reinterpret_cast<const float2*>(w + (size_t)row * E + c0 + khalf);
      a[mi].x = t.x;
      a[mi].y = t.y;
    }
#pragma unroll
    for (int ni = 0; ni < 2; ++ni) {
      const int col = n0 + ni * 16 + l15;   // s
      const float2 t =
          *reinterpret_cast<const float2*>(cb + (size_t)col * E + c0 + khalf);
      bb[ni].x = t.x;
      bb[ni].y = t.y;
    }
#pragma unroll
    for (int mi = 0; mi < 2; ++mi)
#pragma unroll
      for (int ni = 0; ni < 2; ++ni)
        acc[mi][ni] = wmma_f32(a[mi], bb[ni], acc[mi][ni]);
  }

#pragma unroll
  for (int mi = 0; mi < 2; ++mi) {
#pragma unroll
    for (int r = 0; r < 8; ++r) {
      const int orow = m0 + mi * 16 + rhalf + r;
      const float bv = bias[orow];
#pragma unroll
      for (int ni = 0; ni < 2; ++ni) {
        const int col = n0 + ni * 16 + l15;
        out[(size_t)b * E * S + (size_t)orow * S + col] = acc[mi][ni][r] + bv;
      }
    }
  }
}

// -------------------------------------------------------------------------
extern "C" void kernel_launch(void* const* d_in, const int* in_sizes, int n_in,
                              void* d_out, int out_size, void* d_ws,
                              size_t ws_size, hipStream_t stream) {
  (void)in_sizes; (void)n_in; (void)out_size; (void)ws_size;

  const float* q_in = (const float*)d_in[0];
  const float* k_in = (const float*)d_in[1];
  const float* v_in = (const float*)d_in[2];
  const float* wq_w = (const float*)d_in[3];
  const float* wq_b = (const float*)d_in[4];
  const float* wk_w = (const float*)d_in[5];
  const float* wk_b = (const float*)d_in[6];
  const float* wv_w = (const float*)d_in[7];
  const float* wv_b = (const float*)d_in[8];
  const float* fc_w = (const float*)d_in[9];
  const float* fc_b = (const float*)d_in[10];

  float* out   = (float*)d_out;                 // [B, 768, 1024]
  float* probs = out + (size_t)Bn * E * S;      // [B*NH, 1024, 1024]

  float* qws  = (float*)d_ws;                   // [BH, S, DP]  25.2 MB
  float* ktws = qws  + (size_t)Bn * NH * S * DP;  // [BH, DP, S]
  float* vws  = ktws + (size_t)Bn * NH * S * DP;  // [BH, S, DP]
  float* ctxws = qws;  // reuse Q buffer: Q dead once scores are written

  const dim3 blk(128);  // 4 waves; each wave owns one 32x32 output tile

  proj_kernel<<<dim3(192, Bn), blk, 0, stream>>>(q_in, wq_w, wq_b, qws, 0);
  proj_kernel<<<dim3(192, Bn), blk, 0, stream>>>(k_in, wk_w, wk_b, ktws, 1);
  proj_kernel<<<dim3(192, Bn), blk, 0, stream>>>(v_in, wv_w, wv_b, vws, 0);

  scores_kernel<<<dim3(256, Bn * NH), blk, 0, stream>>>(qws, ktws, probs);
  softmax_kernel<<<dim3(Bn * NH * S), dim3(256), 0, stream>>>(probs);
  ctx_kernel<<<dim3(16, Bn * NH), blk, 0, stream>>>(probs, vws, ctxws);
  fc_kernel<<<dim3(192, Bn), blk, 0, stream>>>(ctxws, fc_w, fc_b, out);
}